// RGCN_86431921865314
// MI455X (gfx1250) — compile-verified
//
#include <hip/hip_runtime.h>

#define N_NODES 50000
#define E_EDGES 400000
#define DIM     128
#define NREL    8
#define NBASIS  4
#define EPSV    1e-8f

typedef float v2f __attribute__((ext_vector_type(2)));
typedef float v8f __attribute__((ext_vector_type(8)));

// ---------------------------------------------------------------------------
// Kernel 1: Hb[n][b][:] = h[n] @ V[b]   (b = 0..3, interleaved for gather
// locality) and  agg[n][:] = h[n] @ loop_w + bias  (m == NBASIS pass).
// Grid: (3125 node tiles, 5 matrices), block = 256 (8 waves, wave32).
// Each wave computes one 16x16 fp32 tile with 32 x V_WMMA_F32_16X16X4_F32.
// ---------------------------------------------------------------------------
__global__ __launch_bounds__(256)
void rgcn_basis_gemm(const float* __restrict__ h,
                     const float* __restrict__ V,
                     const float* __restrict__ loop_w,
                     const float* __restrict__ bias,
                     float* __restrict__ Hb,    // [N, NBASIS, D]
                     float* __restrict__ agg)   // [N, D]
{
    const int tile = blockIdx.x;            // 16-node tile
    const int m    = blockIdx.y;            // 0..NBASIS (NBASIS == self loop)
    const int wave = threadIdx.x >> 5;      // col tile 0..7
    const int lane = threadIdx.x & 31;
    const int kh   = lane >> 4;             // half: selects K offset 0 / 2
    const int l15  = lane & 15;
    const int row  = tile * 16 + l15;       // A-row for this lane
    const int colb = wave * 16;
    const int col  = colb + l15;            // B/D column for this lane

    const float* W  = (m < NBASIS) ? (V + (size_t)m * DIM * DIM) : loop_w;
    const float* ap = h + (size_t)row * DIM + 2 * kh;       // A: K = k0+2h, +1
    const float* bp = W + (size_t)(2 * kh) * DIM + col;     // B: rows K, col

    v8f acc = {0.f, 0.f, 0.f, 0.f, 0.f, 0.f, 0.f, 0.f};
    #pragma unroll
    for (int k0 = 0; k0 < DIM; k0 += 4) {
        v2f a = *(const v2f*)ap;            // h[row][k0+2h : k0+2h+1]
        v2f b;
        b.x = bp[0];                        // W[k0+2h    ][col]
        b.y = bp[DIM];                      // W[k0+2h + 1][col]
        acc = __builtin_amdgcn_wmma_f32_16x16x4_f32(
                  false, a, false, b, (short)0, acc, false, false);
        ap += 4;
        bp += 4 * DIM;
    }

    if (m < NBASIS) {
        #pragma unroll
        for (int v = 0; v < 8; ++v) {
            const int node = tile * 16 + v + 8 * kh;        // row M of D tile
            Hb[((size_t)node * NBASIS + m) * DIM + col] = acc[v];
        }
    } else {
        const float bc = bias[col];
        #pragma unroll
        for (int v = 0; v < 8; ++v) {
            const int node = tile * 16 + v + 8 * kh;
            agg[(size_t)node * DIM + col] = acc[v] + bc;
        }
    }
}

// ---------------------------------------------------------------------------
// Kernel 2: per-edge message + scatter-add.  1 wave per edge; lane l owns
// features [4l, 4l+3].  Each edge reads one contiguous 2 KB block of Hb.
// ---------------------------------------------------------------------------
__global__ __launch_bounds__(256)
void rgcn_edge_scatter(const float* __restrict__ Hb,      // [N, NBASIS, D]
                       const float* __restrict__ w_comp,  // [R, NBASIS]
                       const float* __restrict__ norm,
                       const int* __restrict__ src,
                       const int* __restrict__ dst,
                       const int* __restrict__ etype,
                       float* __restrict__ agg)
{
    const int e = blockIdx.x * 8 + (threadIdx.x >> 5);
    if (e >= E_EDGES) return;
    const int lane = threadIdx.x & 31;

    const int   s  = src[e];
    const int   d  = dst[e];
    const int   t  = etype[e];
    const float nr = norm[e];
    const float c0 = w_comp[t * NBASIS + 0];
    const float c1 = w_comp[t * NBASIS + 1];
    const float c2 = w_comp[t * NBASIS + 2];
    const float c3 = w_comp[t * NBASIS + 3];

    const float4* p = (const float4*)(Hb + (size_t)s * NBASIS * DIM);
    const float4 v0 = p[lane];           // basis 0 (32 float4 per basis row)
    const float4 v1 = p[32 + lane];
    const float4 v2 = p[64 + lane];
    const float4 v3 = p[96 + lane];

    float4 msg;
    msg.x = nr * (c0 * v0.x + c1 * v1.x + c2 * v2.x + c3 * v3.x);
    msg.y = nr * (c0 * v0.y + c1 * v1.y + c2 * v2.y + c3 * v3.y);
    msg.z = nr * (c0 * v0.z + c1 * v1.z + c2 * v2.z + c3 * v3.z);
    msg.w = nr * (c0 * v0.w + c1 * v1.w + c2 * v2.w + c3 * v3.w);

    float* ag = agg + (size_t)d * DIM + lane * 4;
    atomicAdd(ag + 0, msg.x);
    atomicAdd(ag + 1, msg.y);
    atomicAdd(ag + 2, msg.z);
    atomicAdd(ag + 3, msg.w);
}

// ---------------------------------------------------------------------------
// Kernel 3: y = relu(agg) @ ffn_w.T  (ReLU fused into the A-operand loads).
// ---------------------------------------------------------------------------
__global__ __launch_bounds__(256)
void rgcn_ffn_gemm(const float* __restrict__ agg,
                   const float* __restrict__ ffn_w,   // [out, in] row-major
                   float* __restrict__ y)             // [N, D]
{
    const int tile = blockIdx.x;
    const int wave = threadIdx.x >> 5;
    const int lane = threadIdx.x & 31;
    const int kh   = lane >> 4;
    const int l15  = lane & 15;
    const int row  = tile * 16 + l15;
    const int colb = wave * 16;
    const int col  = colb + l15;

    const float* ap = agg + (size_t)row * DIM + 2 * kh;
    const float* bp = ffn_w + (size_t)col * DIM + 2 * kh;  // B[k][col] = ffn_w[col][k]

    v8f acc = {0.f, 0.f, 0.f, 0.f, 0.f, 0.f, 0.f, 0.f};
    #pragma unroll
    for (int k0 = 0; k0 < DIM; k0 += 4) {
        v2f a = *(const v2f*)ap;
        a.x = fmaxf(a.x, 0.f);                    // hn = relu(agg)
        a.y = fmaxf(a.y, 0.f);
        v2f b = *(const v2f*)bp;                  // contiguous pair in K
        acc = __builtin_amdgcn_wmma_f32_16x16x4_f32(
                  false, a, false, b, (short)0, acc, false, false);
        ap += 4;
        bp += 4;
    }

    #pragma unroll
    for (int v = 0; v < 8; ++v) {
        const int node = tile * 16 + v + 8 * kh;
        y[(size_t)node * DIM + col] = acc[v];
    }
}

// ---------------------------------------------------------------------------
// Kernel 4: out = LayerNorm(y + ffn_b + h) * ln_g + ln_b.  One 128-thread
// block per node; wave32 shfl_xor reduction + 4-entry LDS cross-wave combine.
// ---------------------------------------------------------------------------
__global__ __launch_bounds__(128)
void rgcn_layernorm(const float* __restrict__ y,
                    const float* __restrict__ h,
                    const float* __restrict__ ffn_b,
                    const float* __restrict__ ln_g,
                    const float* __restrict__ ln_b,
                    float* __restrict__ out)
{
    const int n = blockIdx.x;
    const int t = threadIdx.x;

    const float v = y[(size_t)n * DIM + t] + ffn_b[t] + h[(size_t)n * DIM + t];

    float s1 = v;
    float s2 = v * v;
    #pragma unroll
    for (int o = 16; o >= 1; o >>= 1) {
        s1 += __shfl_xor(s1, o, 32);
        s2 += __shfl_xor(s2, o, 32);
    }

    __shared__ float p1[4], p2[4];
    const int w = t >> 5;
    if ((t & 31) == 0) { p1[w] = s1; p2[w] = s2; }
    __syncthreads();

    const float tot1 = p1[0] + p1[1] + p1[2] + p1[3];
    const float tot2 = p2[0] + p2[1] + p2[2] + p2[3];
    const float mu   = tot1 * (1.0f / DIM);
    const float var  = tot2 * (1.0f / DIM) - mu * mu;
    const float inv  = rsqrtf(var + EPSV);

    out[(size_t)n * DIM + t] = (v - mu) * inv * ln_g[t] + ln_b[t];
}

// ---------------------------------------------------------------------------
extern "C" void kernel_launch(void* const* d_in, const int* in_sizes, int n_in,
                              void* d_out, int out_size, void* d_ws, size_t ws_size,
                              hipStream_t stream)
{
    (void)in_sizes; (void)n_in; (void)out_size; (void)ws_size;

    const float* h      = (const float*)d_in[0];
    const float* V      = (const float*)d_in[1];
    const float* w_comp = (const float*)d_in[2];
    const float* loop_w = (const float*)d_in[3];
    const float* bias   = (const float*)d_in[4];
    const float* ffn_w  = (const float*)d_in[5];
    const float* ffn_b  = (const float*)d_in[6];
    const float* ln_g   = (const float*)d_in[7];
    const float* ln_b   = (const float*)d_in[8];
    const float* norm   = (const float*)d_in[9];
    const int*   src    = (const int*)d_in[10];
    const int*   dst    = (const int*)d_in[11];
    const int*   etype  = (const int*)d_in[12];
    float*       out    = (float*)d_out;

    // Workspace: Hb [N, NBASIS, D] (102.4 MB) | agg [N, D] (25.6 MB).
    // y reuses the Hb region (Hb is dead after the edge pass).
    float* Hb  = (float*)d_ws;
    float* agg = Hb + (size_t)N_NODES * NBASIS * DIM;
    float* yb  = Hb;

    dim3 g1(N_NODES / 16, NBASIS + 1);                  // 3125 x 5, exact
    rgcn_basis_gemm<<<g1, 256, 0, stream>>>(h, V, loop_w, bias, Hb, agg);

    rgcn_edge_scatter<<<E_EDGES / 8, 256, 0, stream>>>( // 50000 blocks, exact
        Hb, w_comp, norm, src, dst, etype, agg);

    rgcn_ffn_gemm<<<N_NODES / 16, 256, 0, stream>>>(agg, ffn_w, yb);

    rgcn_layernorm<<<N_NODES, 128, 0, stream>>>(yb, h, ffn_b, ln_g, ln_b, out);
}